// SNN_conv_74234214744623
// MI455X (gfx1250) — compile-verified
//
#include <hip/hip_runtime.h>

// TTFS spiking conv (3x3, SAME pad with MAX_T) for MI455X / gfx1250.
// One wave32 per neuron; f32 WMMA 16x16x4 triangular matmuls implement the
// blocked inclusive prefix sums over the sorted spike order.

#define MAX_T   100000.0f
#define THETA   1.0f
#define EPSC    1e-10f

typedef __attribute__((ext_vector_type(2))) float v2f;
typedef __attribute__((ext_vector_type(8))) float v8f;

constexpr int Bn = 8, Hh = 32, Ww = 32, Cc = 16, KC = 32;
constexpr int Mreal = 3 * 3 * Cc + 1;   // 145 (incl. bias row)
constexpr int Mpad  = 160;              // 10 blocks of 16
constexpr int WAVES = 8;                // neurons per workgroup (256 threads)

__global__ __launch_bounds__(256) void ttfs_conv_wmma(
    const float* __restrict__ x, const float* __restrict__ w,
    float* __restrict__ y)
{
    __shared__ float wlds[Mpad * KC];        // 20480 B, rows >=145 zeroed
    __shared__ float praw[WAVES][Mpad];      // unsorted spike times
    __shared__ float xs[WAVES][Mpad + 8];    // sorted times (+1 for x_next)
    __shared__ int   sidx[WAVES][Mpad];      // sorted -> original index

    const int tid  = threadIdx.x;
    const int wid  = tid >> 5;
    const int lane = tid & 31;

    // ---- stage weights into LDS (pad rows -> 0 so padded entries are inert)
    for (int i = tid; i < Mreal * KC; i += 256) wlds[i] = w[i];
    for (int i = tid; i < (Mpad - Mreal) * KC; i += 256) wlds[Mreal * KC + i] = 0.0f;

    // ---- neuron coordinates
    const int n  = blockIdx.x * WAVES + wid;
    const int wq = n / Ww;
    const int w0 = n - wq * Ww;
    const int bi = wq / Hh;
    const int h0 = wq - bi * Hh;

    // ---- gather 145 spike times (kh,kw,c order; SAME pad = MAX_T; bias = 1)
    float v[5];
    #pragma unroll
    for (int t = 0; t < 5; ++t) {
        const int m = lane + 32 * t;
        float val = MAX_T;
        if (m < Mreal) {
            if (m == Mreal - 1) {
                val = 1.0f;                          // bias spike
            } else {
                const int kh = m / (3 * Cc);
                const int r  = m - kh * (3 * Cc);
                const int kw = r / Cc;
                const int ci = r - kw * Cc;
                const int hh = h0 + kh - 1, ww2 = w0 + kw - 1;
                if (hh >= 0 && hh < Hh && ww2 >= 0 && ww2 < Ww)
                    val = x[((bi * Hh + hh) * Ww + ww2) * Cc + ci];
            }
            praw[wid][m] = val;
        }
        v[t] = val;
    }
    __syncthreads();

    // ---- stable rank sort (lexicographic on (value, index) == stable argsort)
    int rank[5] = {0, 0, 0, 0, 0};
    for (int j = 0; j < Mreal; ++j) {
        const float pj = praw[wid][j];           // LDS broadcast
        #pragma unroll
        for (int t = 0; t < 5; ++t) {
            const int m = lane + 32 * t;
            rank[t] += (pj < v[t] || (pj == v[t] && j < m)) ? 1 : 0;
        }
    }
    #pragma unroll
    for (int t = 0; t < 5; ++t) {
        const int m = lane + 32 * t;
        if (m < Mreal) { xs[wid][rank[t]] = v[t]; sidx[wid][rank[t]] = m; }
    }
    if (lane <= Mpad - Mreal) {                  // pad entries 145..160
        xs[wid][Mreal + lane] = MAX_T;
        if (lane < Mpad - Mreal) sidx[wid][Mreal + lane] = Mreal; // zero row
    }
    __syncthreads();

    // ---- blocked inclusive prefix sums via triangular f32 WMMA 16x16x4
    const int lm    = lane & 15;                 // A row / B,D column
    const int kb    = (lane < 16) ? 0 : 2;       // K pair held by this lane
    const int rbase = (lane < 16) ? 0 : 8;       // D rows held by this lane

    v2f A[4];                                    // columns [4t+kb, 4t+kb+1] of
    #pragma unroll                               // 16x16 lower-triangular ones
    for (int t = 0; t < 4; ++t) {
        A[t][0] = (4 * t + kb     <= lm) ? 1.0f : 0.0f;
        A[t][1] = (4 * t + kb + 1 <= lm) ? 1.0f : 0.0f;
    }

    float carry_cw[2]  = {0.0f, 0.0f};
    float carry_cwx[2] = {0.0f, 0.0f};
    float acc[2]       = {MAX_T, MAX_T};

    for (int b = 0; b < Mpad / 16; ++b) {
        #pragma unroll
        for (int h = 0; h < 2; ++h) {            // two 16-channel column tiles
            v8f dcw  = {};
            v8f dcwx = {};
            const int c = h * 16 + lm;
            #pragma unroll
            for (int t = 0; t < 4; ++t) {
                const int   k0  = b * 16 + 4 * t + kb;
                const float w0v = wlds[sidx[wid][k0]     * KC + c];
                const float w1v = wlds[sidx[wid][k0 + 1] * KC + c];
                const float x0v = xs[wid][k0];
                const float x1v = xs[wid][k0 + 1];
                v2f bw; bw[0] = w0v;        bw[1] = w1v;
                v2f bx; bx[0] = w0v * x0v;  bx[1] = w1v * x1v;
                dcw  = __builtin_amdgcn_wmma_f32_16x16x4_f32(
                           false, A[t], false, bw, (short)0, dcw,  false, false);
                dcwx = __builtin_amdgcn_wmma_f32_16x16x4_f32(
                           false, A[t], false, bx, (short)0, dcwx, false, false);
            }
            const float ccw  = carry_cw[h];
            const float ccwx = carry_cwx[h];
            #pragma unroll
            for (int r = 0; r < 8; ++r) {        // element-wise TTFS firing test
                const int   k    = b * 16 + rbase + r;
                const float cwv  = dcw[r]  + ccw;
                const float cwxv = dcwx[r] + ccwx;
                const float xk   = xs[wid][k];
                const float xk1  = xs[wid][k + 1];
                const float den  = fminf(fmaxf(cwv - THETA, EPSC), 1e10f);
                const float o    = cwxv / den;
                const bool  ok   = (cwv >= THETA) && (o >= xk) && (o <= xk1);
                acc[h] = fminf(acc[h], ok ? o : MAX_T);
            }
            // carry = inclusive total at row 15 (lanes 16..31, VGPR 7)
            const float tcw  = dcw[7]  + ccw;
            const float tcwx = dcwx[7] + ccwx;
            carry_cw[h]  = __shfl(tcw,  16 + lm, 32);
            carry_cwx[h] = __shfl(tcwx, 16 + lm, 32);
        }
    }

    // ---- min across the two row-halves (lane L <-> L+16 hold same channel)
    #pragma unroll
    for (int h = 0; h < 2; ++h)
        acc[h] = fminf(acc[h], __shfl_xor(acc[h], 16, 32));

    if (lane < 16) {
        y[n * KC + lane]      = acc[0];
        y[n * KC + 16 + lane] = acc[1];
    }
}

extern "C" void kernel_launch(void* const* d_in, const int* in_sizes, int n_in,
                              void* d_out, int out_size, void* d_ws, size_t ws_size,
                              hipStream_t stream) {
    (void)in_sizes; (void)n_in; (void)out_size; (void)d_ws; (void)ws_size;
    const float* x = (const float*)d_in[0];   // [8,32,32,16] f32
    const float* w = (const float*)d_in[1];   // [145,32] f32
    float* y = (float*)d_out;                 // [8,32,32,32] f32
    const int neurons = Bn * Hh * Ww;         // 8192 -> 1024 blocks of 8 waves
    hipLaunchKernelGGL(ttfs_conv_wmma, dim3(neurons / WAVES), dim3(256), 0, stream,
                       x, w, y);
}